// DiagonalPositionalEncoding2D_2310692406011
// MI455X (gfx1250) — compile-verified
//
#include <hip/hip_runtime.h>
#include <math.h>

// ---------------------------------------------------------------------------
// DiagonalPositionalEncoding2D for MI455X (gfx1250).
//
// Output [8,256,256,256] fp32 = 512 MiB of pure stores (input values unused).
// Strategy: each workgroup computes one tile (fixed i, 32 j's x 256 ch =
// 32 KB) of the batch-invariant embedding into LDS ONCE, then the 8 waves of
// the block each issue a single TDM tensor_store_from_lds (contiguous 32 KB)
// to their batch copy -> 8x write replication with 1x compute, using the
// CDNA5 Tensor Data Mover path (TENSORcnt / s_wait_tensorcnt).
// ---------------------------------------------------------------------------

typedef unsigned int v4u __attribute__((ext_vector_type(4)));
typedef int          v4i __attribute__((ext_vector_type(4)));
typedef int          v8i __attribute__((ext_vector_type(8)));

#if defined(__has_builtin)
#if __has_builtin(__builtin_amdgcn_tensor_store_from_lds)
#define USE_TDM 1
#endif
#endif

#define DIMX 256
#define DIMY 256
#define DIMC 256
#define NB   8
#define TJ   32                         // j-rows per tile
#define TILE_FLOATS (TJ * DIMC)         // 8192 floats = 32 KB
// log2(10000)/64  (inv_freq[k] = 10000^(-k/64) = exp2(-k * this))
#define NEG_L2F 0.20762050593046014f

__global__ __launch_bounds__(256) void diag_pe_kernel(float* __restrict__ out) {
  __shared__ float smem[TILE_FLOATS];

  const int tid = threadIdx.x;
  const int i   = blockIdx.x >> 3;   // 0..255  (x index)
  const int jt  = blockIdx.x & 7;    // 0..7    (j tile)
  const float fi = (float)i;

  // 512 (j,kgroup) pairs per tile; 2 per thread. kgroup = 4 consecutive freqs
  // -> one float4 per quarter, 16-B-stride coalesced/conflict-free LDS stores.
#pragma unroll
  for (int it = 0; it < 2; ++it) {
    const int p  = tid + (it << 8);      // 0..511
    const int jj = p >> 4;               // 0..31 (j within tile)
    const int kg = p & 15;               // 0..15 (freq group of 4)
    const float fj = (float)(jt * TJ + jj);
    const float pr = fj - fi;            // anti-diagonal phase base
    const float pl = fj + fi;            // diagonal phase base

    float srv[4], crv[4], slv[4], clv[4];
#pragma unroll
    for (int m = 0; m < 4; ++m) {
      const float k = (float)((kg << 2) + m);
      const float f = exp2f(-NEG_L2F * k);
      sincosf(pr * f, &srv[m], &crv[m]);
      sincosf(pl * f, &slv[m], &clv[m]);
    }

    float4* row = (float4*)&smem[jj * DIMC];   // 64 float4 per row
    row[kg]      = make_float4(srv[0], srv[1], srv[2], srv[3]);  // ch   0..63
    row[16 + kg] = make_float4(crv[0], crv[1], crv[2], crv[3]);  // ch  64..127
    row[32 + kg] = make_float4(slv[0], slv[1], slv[2], slv[3]);  // ch 128..191
    row[48 + kg] = make_float4(clv[0], clv[1], clv[2], clv[3]);  // ch 192..255
  }

  __syncthreads();

#ifdef USE_TDM
  // One TDM store per wave (8 waves -> 8 batch copies). TDM ignores EXEC, so
  // guarding on lane 0 issues exactly one op per wave.
  if ((tid & 31) == 0) {
    const int b = tid >> 5;
    const unsigned long long ga =
        (unsigned long long)(uintptr_t)out +
        4ull * ((unsigned long long)b * (DIMX * DIMY * DIMC) +
                (unsigned long long)i * (DIMY * DIMC) +
                (unsigned long long)(jt * TILE_FLOATS));

    typedef __attribute__((address_space(3))) char lds_char;
    const unsigned lds_base =
        (unsigned)(unsigned long long)(lds_char*)(void*)smem;

    // D# group 0: count=1 | lds_addr | global_addr[31:0] | ga[56:32], type=2
    v4u g0 = { 1u,
               lds_base,
               (unsigned)(ga & 0xffffffffull),
               (unsigned)((ga >> 32) & 0x01ffffffull) | 0x80000000u };
    // D# group 1: 1-D contiguous tile of TILE_FLOATS 4-byte elements.
    //  d0: wg_mask=0, data_size=2 (4B)
    //  d1[31:16]=tensor_dim0[15:0]   d2[15:0]=tensor_dim0[31:16]
    //  d2[31:16]=tensor_dim1[15:0]   d3[31:16]=tile_dim0
    //  d4: tile_dim1=0, tile_dim2=0 (unused)   d5: tensor_dim0_stride
    v8i g1 = { (int)0x00020000,
               (int)((unsigned)TILE_FLOATS << 16),
               (int)(1u << 16),
               (int)((unsigned)TILE_FLOATS << 16),
               0,
               (int)TILE_FLOATS,
               0, 0 };
    v4i g2 = {0, 0, 0, 0};
    v4i g3 = {0, 0, 0, 0};
#if __clang_major__ >= 23
    v8i gz = {0, 0, 0, 0, 0, 0, 0, 0};
    __builtin_amdgcn_tensor_store_from_lds(g0, g1, g2, g3, gz, 0);
#else
    __builtin_amdgcn_tensor_store_from_lds(g0, g1, g2, g3, 0);
#endif
  }
  __builtin_amdgcn_s_wait_tensorcnt(0);
#else
  // Fallback (builtin unavailable / host pass): coalesced float4 stores of the
  // LDS tile to all 8 batch copies.
  const float4* s4 = (const float4*)smem;
  float4* o4 = (float4*)out;
  const size_t tile4  = (size_t)(i * (DIMY * DIMC) + jt * TILE_FLOATS) >> 2;
  const size_t batch4 = (size_t)(DIMX * DIMY * DIMC) >> 2;
  for (int v = tid; v < (TILE_FLOATS >> 2); v += 256) {
    const float4 val = s4[v];
#pragma unroll
    for (int b = 0; b < NB; ++b) {
      o4[(size_t)b * batch4 + tile4 + (size_t)v] = val;
    }
  }
#endif
}

extern "C" void kernel_launch(void* const* d_in, const int* in_sizes, int n_in,
                              void* d_out, int out_size, void* d_ws, size_t ws_size,
                              hipStream_t stream) {
  (void)d_in; (void)in_sizes; (void)n_in; (void)d_ws; (void)ws_size; (void)out_size;
  float* out = (float*)d_out;
  dim3 grid(DIMX * (DIMY / TJ));   // 256 * 8 = 2048 blocks
  dim3 block(256);                 // 8 waves -> 8 batch DMA copies per tile
  diag_pe_kernel<<<grid, block, 0, stream>>>(out);
}